// NeighborGatherLayer3D_50551765074717
// MI455X (gfx1250) — compile-verified
//
#include <hip/hip_runtime.h>
#include <cstdint>
#include <cstddef>

// NeighborGatherLayer3D — pure bandwidth-bound gather/copy.
// out[b,l,k,:,:] = (idx=nbr[l,k]) < 0 ? 0 : in[b,idx,:,:]
// Tile = T*C = 256 floats = 64 float4 = 1KB, contiguous in src and dst.
//
// MI455X path: wave-level async DMA global->LDS->global (ASYNCcnt), 512B per
// instruction per wave, no VGPR round trip; NT stores keep the write-once
// 425MB output out of L2 so the 61MB input (reused ~6x) stays resident.

typedef float f4 __attribute__((ext_vector_type(4)));

#define B_N   32
#define L_N   1855
#define K_N   7
#define TILE4 64            // float4 per (b,l) tile
#define BLK   256           // threads per block (8 waves, wave32)
#define ITERS 8             // 2048 float4 per block / 256 threads

__global__ __launch_bounds__(BLK) void neighbor_gather_async(
    const f4* __restrict__ in,      // (B, L, 64) float4
    const int* __restrict__ nbr,    // (L, K)
    f4* __restrict__ out)           // (B, L, K, 64) float4
{
    __shared__ f4 stage[BLK * ITERS];   // 32 KB staging

    const int l   = blockIdx.x;
    const int k   = blockIdx.y;
    const int tid = threadIdx.x;
    const int idx = nbr[l * K_N + k];   // uniform -> s_load

    // With e = it*256 + tid: f = tid & 63 (invariant), b = 4*it + (tid>>6).
    // => all addresses advance by a constant stride per iteration.
    const int b0 = tid >> 6;            // 0..3
    const int f  = tid & (TILE4 - 1);   // 0..63

    const size_t dstStrideB = (size_t)L_N * K_N * TILE4;  // float4 per batch
    const size_t srcStrideB = (size_t)L_N * TILE4;

    f4* dst = out + ((size_t)l * K_N + k) * TILE4
                  + (size_t)b0 * dstStrideB + f;
    const size_t dstStep = 4 * dstStrideB;                // b += 4 per iter

    if (idx < 0) {
        // Masked neighbor: stream zeros, non-temporal.
        const f4 z = {0.f, 0.f, 0.f, 0.f};
#pragma unroll
        for (int it = 0; it < ITERS; ++it) {
            __builtin_nontemporal_store(z, dst);
            dst += dstStep;
        }
        return;
    }

    const f4* src = in + (size_t)idx * TILE4
                       + (size_t)b0 * srcStrideB + f;
    const size_t srcStep = 4 * srcStrideB;

    unsigned lds = (unsigned)(uintptr_t)(&stage[tid]);    // +4KB per iter

    // Phase 1: async DMA global -> LDS. Input kept regular-temporal
    // (L2-resident, ~6x reuse absorbed on-chip).
#pragma unroll
    for (int it = 0; it < ITERS; ++it) {
        asm volatile("global_load_async_to_lds_b128 %0, %1, off"
                     :: "v"(lds), "v"(src) : "memory");
        src += srcStep;
        lds += BLK * sizeof(f4);
    }

    // Each thread stores exactly the LDS bytes it loaded -> per-wave
    // ASYNCcnt ordering suffices; no workgroup barrier needed.
    asm volatile("s_wait_asynccnt 0" ::: "memory");

    // Phase 2: async DMA LDS -> global, non-temporal.
    lds = (unsigned)(uintptr_t)(&stage[tid]);
#pragma unroll
    for (int it = 0; it < ITERS; ++it) {
        asm volatile("global_store_async_from_lds_b128 %0, %1, off th:TH_STORE_NT"
                     :: "v"(dst), "v"(lds) : "memory");
        dst += dstStep;
        lds += BLK * sizeof(f4);
    }

    // Drain before wave end (in-flight stores read this WG's LDS allocation;
    // S_ENDPGM also implies wait-idle, this is belt-and-braces).
    asm volatile("s_wait_asynccnt 0" ::: "memory");
}

extern "C" void kernel_launch(void* const* d_in, const int* in_sizes, int n_in,
                              void* d_out, int out_size, void* d_ws, size_t ws_size,
                              hipStream_t stream) {
    const f4*  in  = (const f4*)d_in[0];   // inputs (B,L,T,C) f32
    const int* nbr = (const int*)d_in[1];  // neighbor_indices (L,K) i32
    f4*        out = (f4*)d_out;           // (B,L,K,T,C) f32

    dim3 grid(L_N, K_N, 1);
    neighbor_gather_async<<<grid, BLK, 0, stream>>>(in, nbr, out);
}